// ParamNetV2QuantizedCanonicalMotionGenericNorm_50302656971355
// MI455X (gfx1250) — compile-verified
//
#include <hip/hip_runtime.h>
#include <math.h>

// ---------------------------------------------------------------------------
// Sizes
// ---------------------------------------------------------------------------
#define BATCH 4096
#define TSTEPS 64
#define DDIM 64
#define HDIM 128
#define NE 16384

// d_out layout (floats): motion | canonical | diff | ind(int bits)
#define MOTION_OFF 0
#define CAN_OFF (BATCH * 96 * TSTEPS)            // 25165824
#define DIFF_OFF (CAN_OFF + BATCH * 32)          // 25296896
#define IND_OFF (DIFF_OFF + 1)                   // 25296897

// workspace layout (floats)
#define WS_F_OFF 0                               // f: 4096*128
#define WS_CN_OFF (BATCH * HDIM)                 // cnorm: 16384
#define WS_BN_OFF (WS_CN_OFF + NE)               // bn prep: 8 slots * 256
#define WS_DP_OFF (WS_BN_OFF + 8 * 256)          // diff partials: 16
#define WS_CBT_OFF (WS_DP_OFF + 16)              // codebook^T: 16384*128

typedef float v2f __attribute__((ext_vector_type(2)));
typedef float v8f __attribute__((ext_vector_type(8)));

// ---------------------------------------------------------------------------
// Prep: fold BN params into (scale, shift) pairs. 8 slots x 128 ch x 2 floats.
// slot 0: bnc(64) 1: blk_bn0(128) 2: blk_bn1(64) 3: lastbn(64)
// slot 4: tb_bn0[0] 5: tb_bn1[0] 6: tb_bn0[1] 7: tb_bn1[1]  (all 128)
// ---------------------------------------------------------------------------
__global__ void prep_bn_kernel(const float* __restrict__ bnc,
                               const float* __restrict__ blk_bn0,
                               const float* __restrict__ blk_bn1,
                               const float* __restrict__ lastbn,
                               const float* __restrict__ tb_bn0,
                               const float* __restrict__ tb_bn1,
                               float* __restrict__ out)
{
    const float* srcs[8] = {bnc, blk_bn0, blk_bn1, lastbn,
                            tb_bn0, tb_bn1, tb_bn0 + 4 * HDIM, tb_bn1 + 4 * HDIM};
    const int ns[8] = {64, 128, 64, 64, 128, 128, 128, 128};
    int t = threadIdx.x;
    for (int s = 0; s < 8; ++s) {
        int n = ns[s];
        const float* p = srcs[s];
        for (int c = t; c < n; c += blockDim.x) {
            float g = p[c], bb = p[n + c], m = p[2 * n + c], v = p[3 * n + c];
            float sc = g * rsqrtf(v + 1e-5f);
            out[s * 256 + 2 * c] = sc;
            out[s * 256 + 2 * c + 1] = bb - m * sc;
        }
    }
}

// 0.5 * ||c_e||^2 per code
__global__ void prep_cnorm_kernel(const float* __restrict__ codebook,
                                  float* __restrict__ cnorm)
{
    int e = blockIdx.x * blockDim.x + threadIdx.x;
    float s = 0.f;
    for (int h = 0; h < HDIM; ++h) {
        float c = codebook[(size_t)h * NE + e];
        s += c * c;
    }
    cnorm[e] = 0.5f * s;
}

// codebook (H, NE) -> codebookT (NE, H), so VQ B-fragments are contiguous.
__global__ void prep_cbt_kernel(const float* __restrict__ codebook,
                                float* __restrict__ cbT)
{
    int idx = blockIdx.x * blockDim.x + threadIdx.x;  // [0, NE*HDIM)
    int e = idx & (NE - 1);
    int h = idx >> 14;
    cbT[(size_t)e * HDIM + h] = codebook[(size_t)h * NE + e];
}

// ---------------------------------------------------------------------------
// Head: f = lrelu(bn(mtl)) @ Wc^T + bc    (4096x64 -> 4096x128)
// Block = 16 rows, 8 waves, wave w owns column tile w*16. K = 64.
// ---------------------------------------------------------------------------
__global__ __launch_bounds__(256) void head_kernel(
    const float* __restrict__ mtl, const float* __restrict__ Wc,
    const float* __restrict__ bc, const float* __restrict__ bnprep,
    float* __restrict__ fbuf)
{
    __shared__ float fa[16 * DDIM];
    const int tid = threadIdx.x;
    const int lane = tid & 31, wave = tid >> 5;
    const int lm = lane & 15, lh = lane >> 4;
    const int b0 = blockIdx.x * 16;
    const float* act = bnprep + 0 * 256;

    for (int idx = tid; idx < 16 * DDIM; idx += 256) {
        int r = idx >> 6, c = idx & 63;
        float x = mtl[(size_t)(b0 + r) * DDIM + c];
        float a = x * act[2 * c] + act[2 * c + 1];
        fa[idx] = a >= 0.f ? a : 0.2f * a;
    }
    __syncthreads();

    const int n0 = wave * 16;
    float bv = bc[n0 + lm];
    v8f acc = {bv, bv, bv, bv, bv, bv, bv, bv};
    const float* arow = fa + lm * DDIM + 2 * lh;
    const float* brow = Wc + (n0 + lm) * DDIM + 2 * lh;
#pragma unroll
    for (int k4 = 0; k4 < 16; ++k4) {
        v2f a = *(const v2f*)(arow + 4 * k4);
        v2f bf = *(const v2f*)(brow + 4 * k4);
        acc = __builtin_amdgcn_wmma_f32_16x16x4_f32(
            false, a, false, bf, (short)0, acc, false, false);
    }
#pragma unroll
    for (int v = 0; v < 8; ++v) {
        int r = v + 8 * lh;
        fbuf[(size_t)(b0 + r) * HDIM + n0 + lm] = acc[v];
    }
}

// ---------------------------------------------------------------------------
// VQ: per row of f, argmax over codes of (f . c_e - 0.5||c_e||^2).
// Block = 16 rows (A-fragments hoisted into registers for the whole loop),
// 8 waves stride over code tiles, B-fragments are contiguous codebookT rows.
// ---------------------------------------------------------------------------
__global__ __launch_bounds__(256) void vq_kernel(
    const float* __restrict__ fbuf, const float* __restrict__ cbT,
    const float* __restrict__ cnorm, float* __restrict__ out)
{
    __shared__ float fs[16 * HDIM];
    __shared__ float redv[8][16];
    __shared__ int redi[8][16];
    const int tid = threadIdx.x;
    const int lane = tid & 31, wave = tid >> 5;
    const int lm = lane & 15, lh = lane >> 4;
    const int b0 = blockIdx.x * 16;

    // async-stage the 16x128 f tile into LDS (CDNA5 async-to-LDS path)
    {
        const float* g = fbuf + (size_t)b0 * HDIM + tid * 4;
        unsigned lo = (unsigned)(uintptr_t)(fs + tid * 4);
        asm volatile("global_load_async_to_lds_b128 %0, %1, off"
                     :: "v"(lo), "v"(g) : "memory");
        const float* g2 = g + 1024;
        unsigned lo2 = lo + 4096;
        asm volatile("global_load_async_to_lds_b128 %0, %1, off"
                     :: "v"(lo2), "v"(g2) : "memory");
    }
    asm volatile("s_wait_asynccnt 0x0" ::: "memory");
    __syncthreads();

    // hoist A-fragments (invariant over the whole code loop): 32 x v2f
    v2f af[32];
    {
        const float* arow = fs + lm * HDIM + 2 * lh;
#pragma unroll
        for (int k4 = 0; k4 < 32; ++k4)
            af[k4] = *(const v2f*)(arow + 4 * k4);
    }

    float bestv[8];
    int besti[8];
#pragma unroll
    for (int v = 0; v < 8; ++v) { bestv[v] = -INFINITY; besti[v] = 0; }

    for (int e0 = wave * 16; e0 < NE; e0 += 8 * 16) {
        v8f acc = {0.f, 0.f, 0.f, 0.f, 0.f, 0.f, 0.f, 0.f};
        const float* cb = cbT + (size_t)(e0 + lm) * HDIM + 2 * lh;
#pragma unroll
        for (int k4 = 0; k4 < 32; ++k4) {
            v2f bf = *(const v2f*)(cb + 4 * k4);
            acc = __builtin_amdgcn_wmma_f32_16x16x4_f32(
                false, af[k4], false, bf, (short)0, acc, false, false);
        }
        float cn = cnorm[e0 + lm];
        int ei = e0 + lm;
#pragma unroll
        for (int v = 0; v < 8; ++v) {
            float s = acc[v] - cn;
            if (s > bestv[v]) { bestv[v] = s; besti[v] = ei; }
        }
    }

    // reduce across the 16 lanes of each half (rows 0-7 / rows 8-15)
#pragma unroll
    for (int m = 1; m < 16; m <<= 1) {
#pragma unroll
        for (int v = 0; v < 8; ++v) {
            float ov = __shfl_xor(bestv[v], m, 32);
            int oi = __shfl_xor(besti[v], m, 32);
            if (ov > bestv[v] || (ov == bestv[v] && oi < besti[v])) {
                bestv[v] = ov;
                besti[v] = oi;
            }
        }
    }
    if (lm == 0) {
#pragma unroll
        for (int v = 0; v < 8; ++v) {
            redv[wave][v + 8 * lh] = bestv[v];
            redi[wave][v + 8 * lh] = besti[v];
        }
    }
    __syncthreads();
    if (tid < 16) {
        float bv = redv[0][tid];
        int bi = redi[0][tid];
        for (int w = 1; w < 8; ++w) {
            float ov = redv[w][tid];
            int oi = redi[w][tid];
            if (ov > bv || (ov == bv && oi < bi)) { bv = ov; bi = oi; }
        }
        ((int*)out)[IND_OFF + b0 + tid] = bi;
    }
}

// ---------------------------------------------------------------------------
// Canonical tail (scalar, tiny): q gather, diff partial, resblock, projection.
// One thread per batch row; deterministic fixed-order diff reduction.
// ---------------------------------------------------------------------------
__global__ __launch_bounds__(256) void tail_kernel(
    const float* __restrict__ fbuf, const float* __restrict__ cbT,
    const float* __restrict__ bnprep,
    const float* __restrict__ W0, const float* __restrict__ b0v,
    const float* __restrict__ W1, const float* __restrict__ b1v,
    const float* __restrict__ Ws,
    const float* __restrict__ Wo, const float* __restrict__ bo,
    float* __restrict__ out, float* __restrict__ diffp)
{
    __shared__ float red[256];
    int tid = threadIdx.x;
    int b = blockIdx.x * blockDim.x + tid;
    int ind = ((const int*)out)[IND_OFF + b];

    float q[HDIM], a0[HDIM];
    float ds = 0.f;
    for (int c = 0; c < HDIM; ++c) {
        float qc = cbT[(size_t)ind * HDIM + c];
        q[c] = qc;
        float d = qc - fbuf[(size_t)b * HDIM + c];
        ds += d * d;
    }
    red[tid] = ds;
    __syncthreads();
    if (tid == 0) {
        float s = 0.f;
        for (int i = 0; i < 256; ++i) s += red[i];
        diffp[blockIdx.x] = s;
    }

    const float* act0 = bnprep + 1 * 256;   // blk_bn0 (128)
    const float* act1 = bnprep + 2 * 256;   // blk_bn1 (64)
    const float* act2 = bnprep + 3 * 256;   // lastbn  (64)

    for (int c = 0; c < HDIM; ++c) {
        float t = q[c] * act0[2 * c] + act0[2 * c + 1];
        a0[c] = t >= 0.f ? t : 0.2f * t;
    }
    float z[64];
    for (int o = 0; o < 64; ++o) {              // net = a0 @ W0^T + b0
        float acc = b0v[o];
        for (int c = 0; c < HDIM; ++c) acc += a0[c] * W0[o * HDIM + c];
        z[o] = acc;
    }
    float a1[64];
    for (int c = 0; c < 64; ++c) {
        float t = z[c] * act1[2 * c] + act1[2 * c + 1];
        a1[c] = t >= 0.f ? t : 0.2f * t;
    }
    float zz[64];
    for (int o = 0; o < 64; ++o) {              // shortcut + net2
        float acc = b1v[o];
        for (int c = 0; c < 64; ++c) acc += a1[c] * W1[o * 64 + c];
        float xs = 0.f;
        for (int c = 0; c < HDIM; ++c) xs += q[c] * Ws[o * HDIM + c];
        zz[o] = xs + acc;
    }
    float a2[64];
    for (int c = 0; c < 64; ++c) {
        float t = zz[c] * act2[2 * c] + act2[2 * c + 1];
        a2[c] = t >= 0.f ? t : 0.2f * t;
    }
    for (int o = 0; o < 32; ++o) {
        float acc = bo[o];
        for (int c = 0; c < 64; ++c) acc += a2[c] * Wo[o * 64 + c];
        out[CAN_OFF + (size_t)b * 32 + o] = acc;
    }
}

__global__ void diff_final_kernel(const float* __restrict__ diffp,
                                  float* __restrict__ out)
{
    float s = 0.f;
    for (int i = 0; i < BATCH / 256; ++i) s += diffp[i];
    out[DIFF_OFF] = s * (1.f / ((float)BATCH * (float)HDIM));
}

// ---------------------------------------------------------------------------
// Temporal branch, fully fused per batch element:
//   x = in @ W1t^T + b1t ; 2x resblock ; motion = lrelu(x) @ Wm^T + bm
// One workgroup per b: 64 t-rows x 128 ch in LDS, ping-pong two buffers.
// 8 waves; wave w owns column tile w*16 -> its 32 B-fragments are loaded
// once per layer into registers and reused over all 4 row tiles.
// ---------------------------------------------------------------------------
__device__ __forceinline__ void layer_gemm(
    const float* __restrict__ src, float* __restrict__ dst,
    const float* __restrict__ W, const float* __restrict__ bias,
    const float* __restrict__ actp, int mode, int residual,
    int n0, int lm, int lh)
{
    v2f bw[32];
#pragma unroll
    for (int k4 = 0; k4 < 32; ++k4)
        bw[k4] = *(const v2f*)(W + (n0 + lm) * HDIM + 2 * lh + 4 * k4);
    float bv = bias[n0 + lm];
    for (int rt = 0; rt < 4; ++rt) {
        v8f acc = {bv, bv, bv, bv, bv, bv, bv, bv};
        const float* arow = src + (rt * 16 + lm) * HDIM + 2 * lh;
#pragma unroll
        for (int k4 = 0; k4 < 32; ++k4) {
            v2f a = *(const v2f*)(arow + 4 * k4);
            if (mode == 1) {
                float4 p = *(const float4*)(actp + (4 * k4 + 2 * lh) * 2);
                a.x = a.x * p.x + p.y;
                a.y = a.y * p.z + p.w;
            }
            if (mode != 0) {
                a.x = a.x >= 0.f ? a.x : 0.2f * a.x;
                a.y = a.y >= 0.f ? a.y : 0.2f * a.y;
            }
            acc = __builtin_amdgcn_wmma_f32_16x16x4_f32(
                false, a, false, bw[k4], (short)0, acc, false, false);
        }
#pragma unroll
        for (int v = 0; v < 8; ++v) {
            int off = (rt * 16 + v + 8 * lh) * HDIM + n0 + lm;
            dst[off] = residual ? dst[off] + acc[v] : acc[v];
        }
    }
}

__global__ __launch_bounds__(256) void temporal_kernel(
    const float* __restrict__ inputs,
    const float* __restrict__ W1t, const float* __restrict__ b1t,
    const float* __restrict__ tbW0, const float* __restrict__ tbB0,
    const float* __restrict__ tbW1, const float* __restrict__ tbB1,
    const float* __restrict__ Wm, const float* __restrict__ bm,
    const float* __restrict__ bnprep,
    float* __restrict__ out)
{
    __shared__ float bufX[TSTEPS * HDIM];  // 32 KB : residual stream "x"
    __shared__ float bufH[TSTEPS * HDIM];  // 32 KB : intermediate "h"
    const int tid = threadIdx.x;
    const int lane = tid & 31, wave = tid >> 5;
    const int lm = lane & 15, lh = lane >> 4;
    const int b = blockIdx.x;
    const float* in_b = inputs + (size_t)b * (128 * TSTEPS);
    const int n0 = wave * 16;

    // async-stage inputs transposed into LDS: bufH[t*128 + c] = in[b, c, t]
    for (int idx = tid; idx < 128 * TSTEPS; idx += 256) {
        int c = idx >> 6, t = idx & 63;
        const float* g = in_b + idx;
        unsigned lo = (unsigned)(uintptr_t)(bufH + t * HDIM + c);
        asm volatile("global_load_async_to_lds_b32 %0, %1, off"
                     :: "v"(lo), "v"(g) : "memory");
    }
    asm volatile("s_wait_asynccnt 0x0" ::: "memory");
    __syncthreads();

    // L0: bufX = bufH @ W1t^T + b1t  (identity activation)
    layer_gemm(bufH, bufX, W1t, b1t, nullptr, 0, 0, n0, lm, lh);
    __syncthreads();

    for (int blk = 0; blk < 2; ++blk) {
        const float* W0 = tbW0 + (size_t)blk * HDIM * HDIM;
        const float* B0 = tbB0 + blk * HDIM;
        const float* W1 = tbW1 + (size_t)blk * HDIM * HDIM;
        const float* B1 = tbB1 + blk * HDIM;
        const float* act0 = bnprep + (4 + 2 * blk) * 256;  // tb_bn0[blk]
        const float* act1 = bnprep + (5 + 2 * blk) * 256;  // tb_bn1[blk]

        // h = lrelu(bn0(x)) @ W0^T + b0
        layer_gemm(bufX, bufH, W0, B0, act0, 1, 0, n0, lm, lh);
        __syncthreads();
        // x += lrelu(bn1(h)) @ W1^T + b1
        layer_gemm(bufH, bufX, W1, B1, act1, 1, 1, n0, lm, lh);
        __syncthreads();
    }

    // motion = lrelu(x) @ Wm^T + bm ; out[b, o, t] with o in [0,96)
    if (wave < 6) {
        v2f bw[32];
#pragma unroll
        for (int k4 = 0; k4 < 32; ++k4)
            bw[k4] = *(const v2f*)(Wm + (n0 + lm) * HDIM + 2 * lh + 4 * k4);
        float bv = bm[n0 + lm];
        for (int rt = 0; rt < 4; ++rt) {
            v8f acc = {bv, bv, bv, bv, bv, bv, bv, bv};
            const float* arow = bufX + (rt * 16 + lm) * HDIM + 2 * lh;
#pragma unroll
            for (int k4 = 0; k4 < 32; ++k4) {
                v2f a = *(const v2f*)(arow + 4 * k4);
                a.x = a.x >= 0.f ? a.x : 0.2f * a.x;
                a.y = a.y >= 0.f ? a.y : 0.2f * a.y;
                acc = __builtin_amdgcn_wmma_f32_16x16x4_f32(
                    false, a, false, bw[k4], (short)0, acc, false, false);
            }
#pragma unroll
            for (int v = 0; v < 8; ++v) {
                int t = rt * 16 + v + 8 * lh;
                out[MOTION_OFF + (size_t)b * (96 * TSTEPS) +
                    (size_t)(n0 + lm) * TSTEPS + t] = acc[v];
            }
        }
    }
}

// ---------------------------------------------------------------------------
extern "C" void kernel_launch(void* const* d_in, const int* in_sizes, int n_in,
                              void* d_out, int out_size, void* d_ws, size_t ws_size,
                              hipStream_t stream)
{
    const float* inputs   = (const float*)d_in[0];
    const float* mtl      = (const float*)d_in[1];
    const float* bnc      = (const float*)d_in[2];
    const float* Wc       = (const float*)d_in[3];
    const float* bc       = (const float*)d_in[4];
    const float* codebook = (const float*)d_in[5];
    const float* blk_bn0  = (const float*)d_in[6];
    const float* blk_W0   = (const float*)d_in[7];
    const float* blk_b0   = (const float*)d_in[8];
    const float* blk_bn1  = (const float*)d_in[9];
    const float* blk_W1   = (const float*)d_in[10];
    const float* blk_b1   = (const float*)d_in[11];
    const float* blk_Ws   = (const float*)d_in[12];
    const float* lastbn   = (const float*)d_in[13];
    const float* Wo       = (const float*)d_in[14];
    const float* bo       = (const float*)d_in[15];
    const float* W1t      = (const float*)d_in[16];
    const float* b1t      = (const float*)d_in[17];
    const float* tb_bn0   = (const float*)d_in[18];
    const float* tb_W0    = (const float*)d_in[19];
    const float* tb_b0    = (const float*)d_in[20];
    const float* tb_bn1   = (const float*)d_in[21];
    const float* tb_W1    = (const float*)d_in[22];
    const float* tb_b1    = (const float*)d_in[23];
    const float* Wm       = (const float*)d_in[24];
    const float* bm       = (const float*)d_in[25];

    float* out = (float*)d_out;
    float* ws = (float*)d_ws;
    float* wsF = ws + WS_F_OFF;
    float* wsCn = ws + WS_CN_OFF;
    float* wsBn = ws + WS_BN_OFF;
    float* wsDp = ws + WS_DP_OFF;
    float* wsCbT = ws + WS_CBT_OFF;

    prep_bn_kernel<<<1, 256, 0, stream>>>(bnc, blk_bn0, blk_bn1, lastbn,
                                          tb_bn0, tb_bn1, wsBn);
    prep_cnorm_kernel<<<NE / 256, 256, 0, stream>>>(codebook, wsCn);
    prep_cbt_kernel<<<(NE * HDIM) / 256, 256, 0, stream>>>(codebook, wsCbT);

    head_kernel<<<BATCH / 16, 256, 0, stream>>>(mtl, Wc, bc, wsBn, wsF);
    vq_kernel<<<BATCH / 16, 256, 0, stream>>>(wsF, wsCbT, wsCn, out);
    tail_kernel<<<BATCH / 256, 256, 0, stream>>>(wsF, wsCbT, wsBn,
                                                 blk_W0, blk_b0, blk_W1, blk_b1,
                                                 blk_Ws, Wo, bo, out, wsDp);
    diff_final_kernel<<<1, 1, 0, stream>>>(wsDp, out);
    temporal_kernel<<<BATCH, 256, 0, stream>>>(inputs, W1t, b1t,
                                               tb_W0, tb_b0, tb_W1, tb_b1,
                                               Wm, bm, wsBn, out);
}